// BaseGnn_44289702756626
// MI455X (gfx1250) — compile-verified
//
#include <hip/hip_runtime.h>

typedef float v2f __attribute__((ext_vector_type(2)));
typedef float v8f __attribute__((ext_vector_type(8)));

#define N_NODES 50000
#define N_EDGES 600000
#define IN_DIM  128
#define LATENT  128
#define OUT_DIM 40

// ---------------- utility kernels ----------------

__global__ void zero_f32_kernel(float* __restrict__ p, int n) {
    int i = blockIdx.x * blockDim.x + threadIdx.x;
    if (i < n) p[i] = 0.0f;
}

__global__ void degree_kernel(const int* __restrict__ adj, float* __restrict__ deg) {
    int e = blockIdx.x * blockDim.x + threadIdx.x;
    if (e < N_EDGES) {
        int to = adj[2 * e + 1];
        __hip_atomic_fetch_add(&deg[to], 1.0f, __ATOMIC_RELAXED, __HIP_MEMORY_SCOPE_AGENT);
    }
}

__global__ void invdeg_kernel(const float* __restrict__ deg, float* __restrict__ inv) {
    int i = blockIdx.x * blockDim.x + threadIdx.x;
    if (i < N_NODES) inv[i] = 1.0f / fmaxf(deg[i], 1.0f);
}

// One edge handled by 32 threads, 4 floats each (float4 gather, 4 f32 atomics).
__global__ void scatter_kernel(const int* __restrict__ adj,
                               const float* __restrict__ h,
                               float* __restrict__ msg) {
    long long idx = (long long)blockIdx.x * blockDim.x + threadIdx.x;
    int e = (int)(idx >> 5);
    int f = (int)(idx & 31) << 2;
    if (e < N_EDGES) {
        int fro = adj[2 * e];
        int to  = adj[2 * e + 1];
        const float4 v = *(const float4*)(h + (size_t)fro * LATENT + f);
        float* dst = msg + (size_t)to * LATENT + f;
        __hip_atomic_fetch_add(dst + 0, v.x, __ATOMIC_RELAXED, __HIP_MEMORY_SCOPE_AGENT);
        __hip_atomic_fetch_add(dst + 1, v.y, __ATOMIC_RELAXED, __HIP_MEMORY_SCOPE_AGENT);
        __hip_atomic_fetch_add(dst + 2, v.z, __ATOMIC_RELAXED, __HIP_MEMORY_SCOPE_AGENT);
        __hip_atomic_fetch_add(dst + 3, v.w, __ATOMIC_RELAXED, __HIP_MEMORY_SCOPE_AGENT);
    }
}

// ---------------- WMMA f32 16x16x4 GEMMs ----------------
// A 16x4 f32 fragment (2 VGPR/lane): lanes 0-15 -> K=k,k+1 ; lanes 16-31 -> K=k+2,k+3 ; M=lane&15
// B 4x16 f32 fragment: same K split; N=lane&15
// C/D 16x16 f32: c[i] -> row = i + (lane>=16 ? 8 : 0), col = lane&15

// Y[50000x128] = relu(X[50000x128] @ W[128x128] + bias). One block = 16 rows, 8 waves = 8 col tiles.
__global__ __launch_bounds__(256) void gemm_relu_in(const float* __restrict__ X,
                                                    const float* __restrict__ W,
                                                    const float* __restrict__ bias,
                                                    float* __restrict__ Y) {
    __shared__ float Xs[16 * 128];
    const int tid = threadIdx.x;
    const int row_base = blockIdx.x * 16;
    const float* xsrc = X + (size_t)row_base * 128;   // 16 rows are contiguous: 2048 floats
    for (int i = tid; i < 2048; i += 256) Xs[i] = xsrc[i];
    __syncthreads();

    const int lane = tid & 31;
    const int col_base = (tid >> 5) * 16;
    const int mn = lane & 15;
    const int koff = (lane >> 4) << 1;   // 0 or 2

    v8f acc = {};
    for (int k = 0; k < 128; k += 4) {
        v2f a, b;
        a[0] = Xs[mn * 128 + k + koff];
        a[1] = Xs[mn * 128 + k + koff + 1];
        b[0] = W[(size_t)(k + koff) * 128 + col_base + mn];
        b[1] = W[(size_t)(k + koff + 1) * 128 + col_base + mn];
        acc = __builtin_amdgcn_wmma_f32_16x16x4_f32(false, a, false, b,
                                                    (short)0, acc, false, false);
    }
    const float bb = bias[col_base + mn];
    const int rhalf = (lane >> 4) << 3;  // 0 or 8
#pragma unroll
    for (int i = 0; i < 8; i++) {
        float v = fmaxf(acc[i] + bb, 0.0f);
        Y[(size_t)(row_base + i + rhalf) * 128 + col_base + mn] = v;
    }
}

// Y = relu(concat(h, msg*inv_deg) @ Wc[256x128] + bias), done as two K=128 passes.
__global__ __launch_bounds__(256) void gemm_combine_relu(const float* __restrict__ h,
                                                         const float* __restrict__ msg,
                                                         const float* __restrict__ invd,
                                                         const float* __restrict__ Wc,
                                                         const float* __restrict__ bias,
                                                         float* __restrict__ Y) {
    __shared__ float Hs[16 * 128];
    __shared__ float Ms[16 * 128];
    const int tid = threadIdx.x;
    const int row_base = blockIdx.x * 16;
    const float* hsrc = h + (size_t)row_base * 128;
    const float* msrc = msg + (size_t)row_base * 128;
    for (int i = tid; i < 2048; i += 256) {
        Hs[i] = hsrc[i];
        Ms[i] = msrc[i] * invd[row_base + (i >> 7)];   // fold inv_deg into msg tile
    }
    __syncthreads();

    const int lane = tid & 31;
    const int col_base = (tid >> 5) * 16;
    const int mn = lane & 15;
    const int koff = (lane >> 4) << 1;

    v8f acc = {};
    for (int k = 0; k < 128; k += 4) {               // K = 0..127 : h part
        v2f a, b;
        a[0] = Hs[mn * 128 + k + koff];
        a[1] = Hs[mn * 128 + k + koff + 1];
        b[0] = Wc[(size_t)(k + koff) * 128 + col_base + mn];
        b[1] = Wc[(size_t)(k + koff + 1) * 128 + col_base + mn];
        acc = __builtin_amdgcn_wmma_f32_16x16x4_f32(false, a, false, b,
                                                    (short)0, acc, false, false);
    }
    for (int k = 0; k < 128; k += 4) {               // K = 128..255 : msg part
        v2f a, b;
        a[0] = Ms[mn * 128 + k + koff];
        a[1] = Ms[mn * 128 + k + koff + 1];
        b[0] = Wc[(size_t)(128 + k + koff) * 128 + col_base + mn];
        b[1] = Wc[(size_t)(128 + k + koff + 1) * 128 + col_base + mn];
        acc = __builtin_amdgcn_wmma_f32_16x16x4_f32(false, a, false, b,
                                                    (short)0, acc, false, false);
    }
    const float bb = bias[col_base + mn];
    const int rhalf = (lane >> 4) << 3;
#pragma unroll
    for (int i = 0; i < 8; i++) {
        float v = fmaxf(acc[i] + bb, 0.0f);
        Y[(size_t)(row_base + i + rhalf) * 128 + col_base + mn] = v;
    }
}

// out[50000x40] = h @ W_out[128x40] + b_out. 3 waves cover cols 0..47 with guards.
__global__ __launch_bounds__(96) void gemm_out(const float* __restrict__ h,
                                               const float* __restrict__ Wo,
                                               const float* __restrict__ bo,
                                               float* __restrict__ out) {
    __shared__ float Hs[16 * 128];
    const int tid = threadIdx.x;
    const int row_base = blockIdx.x * 16;
    const float* hsrc = h + (size_t)row_base * 128;
    for (int i = tid; i < 2048; i += 96) Hs[i] = hsrc[i];
    __syncthreads();

    const int lane = tid & 31;
    const int col_base = (tid >> 5) * 16;
    const int mn = lane & 15;
    const int koff = (lane >> 4) << 1;
    const int col = col_base + mn;
    const bool ok = (col < OUT_DIM);

    v8f acc = {};
    for (int k = 0; k < 128; k += 4) {
        v2f a, b;
        a[0] = Hs[mn * 128 + k + koff];
        a[1] = Hs[mn * 128 + k + koff + 1];
        b[0] = ok ? Wo[(size_t)(k + koff) * OUT_DIM + col] : 0.0f;
        b[1] = ok ? Wo[(size_t)(k + koff + 1) * OUT_DIM + col] : 0.0f;
        acc = __builtin_amdgcn_wmma_f32_16x16x4_f32(false, a, false, b,
                                                    (short)0, acc, false, false);
    }
    if (ok) {
        const float bb = bo[col];
        const int rhalf = (lane >> 4) << 3;
#pragma unroll
        for (int i = 0; i < 8; i++)
            out[(size_t)(row_base + i + rhalf) * OUT_DIM + col] = acc[i] + bb;
    }
}

// ---------------- launcher ----------------

extern "C" void kernel_launch(void* const* d_in, const int* in_sizes, int n_in,
                              void* d_out, int out_size, void* d_ws, size_t ws_size,
                              hipStream_t stream) {
    const float* X      = (const float*)d_in[0];
    const int*   adj    = (const int*)d_in[1];
    const float* W_in   = (const float*)d_in[2];
    const float* b_in   = (const float*)d_in[3];
    const float* W_comb = (const float*)d_in[4];   // [2, 256, 128]
    const float* b_comb = (const float*)d_in[5];   // [2, 128]
    const float* W_out  = (const float*)d_in[6];   // [128, 40]
    const float* b_out  = (const float*)d_in[7];
    float* out = (float*)d_out;

    char* ws = (char*)d_ws;
    const size_t HB = (size_t)N_NODES * LATENT * sizeof(float);   // 25.6 MB
    float* h0   = (float*)(ws);
    float* h1   = (float*)(ws + HB);
    float* msg  = (float*)(ws + 2 * HB);
    float* deg  = (float*)(ws + 3 * HB);
    float* invd = (float*)(ws + 3 * HB + ((size_t)N_NODES * sizeof(float) + 255 & ~(size_t)255));

    const int NFEAT = N_NODES * LATENT;                 // 6,400,000
    const int ZB_NODE = (N_NODES + 255) / 256;          // zero/invdeg grid
    const int ZB_FEAT = (NFEAT + 255) / 256;            // msg zero grid
    const int EB      = (N_EDGES + 255) / 256;          // degree grid
    const int SB      = (int)(((long long)N_EDGES * 32 + 255) / 256);  // scatter grid
    const int RT      = N_NODES / 16;                   // 3125 row tiles

    // degree + inverse degree
    zero_f32_kernel<<<ZB_NODE, 256, 0, stream>>>(deg, N_NODES);
    degree_kernel<<<EB, 256, 0, stream>>>(adj, deg);
    invdeg_kernel<<<ZB_NODE, 256, 0, stream>>>(deg, invd);

    // h0 = relu(X @ W_in + b_in)
    gemm_relu_in<<<RT, 256, 0, stream>>>(X, W_in, b_in, h0);

    // layer 0
    zero_f32_kernel<<<ZB_FEAT, 256, 0, stream>>>(msg, NFEAT);
    scatter_kernel<<<SB, 256, 0, stream>>>(adj, h0, msg);
    gemm_combine_relu<<<RT, 256, 0, stream>>>(h0, msg, invd,
                                              W_comb, b_comb, h1);

    // layer 1
    zero_f32_kernel<<<ZB_FEAT, 256, 0, stream>>>(msg, NFEAT);
    scatter_kernel<<<SB, 256, 0, stream>>>(adj, h1, msg);
    gemm_combine_relu<<<RT, 256, 0, stream>>>(h1, msg, invd,
                                              W_comb + (size_t)256 * 128, b_comb + 128, h0);

    // out = h0 @ W_out + b_out
    gemm_out<<<RT, 96, 0, stream>>>(h0, W_out, b_out, out);
}